// Smpl_43671227466160
// MI455X (gfx1250) — compile-verified
//
#include <hip/hip_runtime.h>
#include <hip/hip_bf16.h>

typedef __attribute__((ext_vector_type(2))) float v2f;
typedef __attribute__((ext_vector_type(8))) float v8f;

#define NBATCH 512
#define NVERT  6890
#define NJ     24
#define NBETA  10
#define NPF    207            // 9*(J-1)
#define NCOL   (NVERT*3)      // 20670

__device__ const int g_parents[NJ] = {-1,0,0,0,1,2,3,4,5,6,7,8,9,9,9,12,13,14,16,17,18,19,20,21};

// ---------------- K1: rodrigues -> rot_mats [B][24][9], pose_feature [B][207]
__global__ void k_rodrigues(const float* __restrict__ body_pose,
                            const float* __restrict__ global_orient,
                            float* __restrict__ rot,
                            float* __restrict__ pf)
{
    int b = blockIdx.x;
    int j = threadIdx.x;
    if (j >= NJ) return;
    float rx, ry, rz;
    if (j == 0) { rx = global_orient[b*3+0]; ry = global_orient[b*3+1]; rz = global_orient[b*3+2]; }
    else { int o = b*(NJ-1)*3 + (j-1)*3; rx = body_pose[o]; ry = body_pose[o+1]; rz = body_pose[o+2]; }
    float px = rx + 1e-8f, py = ry + 1e-8f, pz = rz + 1e-8f;
    float ang = sqrtf(px*px + py*py + pz*pz);
    float inv = 1.0f / ang;
    float ax = rx*inv, ay = ry*inv, az = rz*inv;
    float s = sinf(ang), c = cosf(ang);
    float k[9] = {0.f, -az, ay,  az, 0.f, -ax,  -ay, ax, 0.f};
    float kk[9];
    #pragma unroll
    for (int m = 0; m < 3; ++m)
        #pragma unroll
        for (int n = 0; n < 3; ++n)
            kk[m*3+n] = k[m*3+0]*k[0*3+n] + k[m*3+1]*k[1*3+n] + k[m*3+2]*k[2*3+n];
    float r[9];
    #pragma unroll
    for (int i = 0; i < 9; ++i) {
        float id = (i==0||i==4||i==8) ? 1.f : 0.f;
        r[i] = id + s*k[i] + (1.f-c)*kk[i];
    }
    #pragma unroll
    for (int i = 0; i < 9; ++i) rot[(b*NJ + j)*9 + i] = r[i];
    if (j >= 1) {
        #pragma unroll
        for (int i = 0; i < 9; ++i) {
            float id = (i==0||i==4||i==8) ? 1.f : 0.f;
            pf[b*NPF + (j-1)*9 + i] = r[i] - id;
        }
    }
}

// ---------------- K2: v_shaped = v_template + shapedirs . betas   [B][V][3]
__global__ void k_vshaped(const float* __restrict__ betas,
                          const float* __restrict__ v_template,
                          const float* __restrict__ shapedirs,
                          float* __restrict__ vsh)
{
    __shared__ float sb[NBETA];
    int b = blockIdx.y;
    if (threadIdx.x < NBETA) sb[threadIdx.x] = betas[b*NBETA + threadIdx.x];
    __syncthreads();
    int v = blockIdx.x*blockDim.x + threadIdx.x;
    if (v >= NVERT) return;
    #pragma unroll
    for (int kd = 0; kd < 3; ++kd) {
        float acc = v_template[v*3 + kd];
        const float* sd = shapedirs + (v*3 + kd)*NBETA;
        #pragma unroll
        for (int l = 0; l < NBETA; ++l) acc += sb[l]*sd[l];
        vsh[(b*NVERT + v)*3 + kd] = acc;
    }
}

// ---------------- K3: Jrest = J_regressor @ v_shaped  [B][24][3]
__global__ void __launch_bounds__(256)
k_jrest(const float* __restrict__ Jreg,
        const float* __restrict__ vsh,
        float* __restrict__ jrest)
{
    __shared__ float red[NJ*3];
    int b = blockIdx.x;
    int tid = threadIdx.x;
    if (tid < NJ*3) red[tid] = 0.f;
    __syncthreads();
    float acc[NJ*3];
    #pragma unroll
    for (int i = 0; i < NJ*3; ++i) acc[i] = 0.f;
    for (int v = tid; v < NVERT; v += blockDim.x) {
        float x = vsh[(b*NVERT+v)*3+0];
        float y = vsh[(b*NVERT+v)*3+1];
        float z = vsh[(b*NVERT+v)*3+2];
        #pragma unroll
        for (int j = 0; j < NJ; ++j) {
            float w = Jreg[j*NVERT + v];
            acc[j*3+0] += w*x; acc[j*3+1] += w*y; acc[j*3+2] += w*z;
        }
    }
    #pragma unroll
    for (int i = 0; i < NJ*3; ++i) {
        float s = acc[i];
        for (int off = 16; off > 0; off >>= 1) s += __shfl_down(s, off, 32);
        if ((tid & 31) == 0) atomicAdd(&red[i], s);
    }
    __syncthreads();
    if (tid < NJ*3) jrest[b*NJ*3 + tid] = red[tid];
}

// ---------------- K4: kinematic chain -> A [B][24][12], joints out
__global__ void k_chain(const float* __restrict__ rot,
                        const float* __restrict__ jrest,
                        const float* __restrict__ transl,
                        float* __restrict__ Aout,
                        float* __restrict__ joints_out)
{
    __shared__ float L[NJ][12];
    __shared__ float G[NJ][12];
    __shared__ float jr[NJ][3];
    int b = blockIdx.x;
    int j = threadIdx.x;
    if (j < NJ) {
        jr[j][0] = jrest[(b*NJ+j)*3+0];
        jr[j][1] = jrest[(b*NJ+j)*3+1];
        jr[j][2] = jrest[(b*NJ+j)*3+2];
    }
    __syncthreads();
    if (j < NJ) {
        int p = g_parents[j];
        float relx = jr[j][0], rely = jr[j][1], relz = jr[j][2];
        if (p >= 0) { relx -= jr[p][0]; rely -= jr[p][1]; relz -= jr[p][2]; }
        #pragma unroll
        for (int m = 0; m < 3; ++m)
            #pragma unroll
            for (int n = 0; n < 3; ++n)
                L[j][m*4+n] = rot[(b*NJ+j)*9 + m*3+n];
        L[j][3] = relx; L[j][7] = rely; L[j][11] = relz;
    }
    __syncthreads();
    if (j == 0) {
        for (int i = 0; i < 12; ++i) G[0][i] = L[0][i];
        for (int i = 1; i < NJ; ++i) {
            int p = g_parents[i];
            for (int m = 0; m < 3; ++m) {
                for (int n = 0; n < 3; ++n)
                    G[i][m*4+n] = G[p][m*4+0]*L[i][0+n] + G[p][m*4+1]*L[i][4+n] + G[p][m*4+2]*L[i][8+n];
                G[i][m*4+3] = G[p][m*4+0]*L[i][3] + G[p][m*4+1]*L[i][7] + G[p][m*4+2]*L[i][11] + G[p][m*4+3];
            }
        }
    }
    __syncthreads();
    if (j < NJ) {
        joints_out[(b*NJ+j)*3+0] = G[j][3]  + transl[b*3+0];
        joints_out[(b*NJ+j)*3+1] = G[j][7]  + transl[b*3+1];
        joints_out[(b*NJ+j)*3+2] = G[j][11] + transl[b*3+2];
        float jx = jr[j][0], jy = jr[j][1], jz = jr[j][2];
        #pragma unroll
        for (int m = 0; m < 3; ++m) {
            float corr = G[j][m*4+0]*jx + G[j][m*4+1]*jy + G[j][m*4+2]*jz;
            Aout[(b*NJ+j)*12 + m*4+0] = G[j][m*4+0];
            Aout[(b*NJ+j)*12 + m*4+1] = G[j][m*4+1];
            Aout[(b*NJ+j)*12 + m*4+2] = G[j][m*4+2];
            Aout[(b*NJ+j)*12 + m*4+3] = G[j][m*4+3] - corr;
        }
    }
}

// ---------------- K5: WMMA fp32 GEMM: vsh += pose_feature @ posedirs (in place)
// M=512 batches (32 tiles of 16), N=20670 (1292 tiles of 16), K=207 pad 208
__global__ void __launch_bounds__(128)
k_poseblend(const float* __restrict__ pf,
            const float* __restrict__ posedirs,
            float* __restrict__ vsh)
{
    __shared__ float As[16][208];
    int mtile = blockIdx.y;
    int tid = threadIdx.x;
    for (int idx = tid; idx < 16*208; idx += 128) {
        int r = idx / 208, c = idx % 208;
        As[r][c] = (c < NPF) ? pf[(mtile*16 + r)*NPF + c] : 0.f;
    }
    __syncthreads();
    int wave = tid >> 5;
    int lane = tid & 31;
    int half = lane >> 4;         // K sub-pair select (ISA 16x4 fp32 A layout)
    int l16  = lane & 15;
    int ntile = blockIdx.x*4 + wave;
    int n  = ntile*16 + l16;
    int nc = n < NCOL ? n : NCOL-1;   // clamp: dup loads, stores guarded later
    int kb = half*2;
    v8f acc = {};
    for (int k = 0; k < 208; k += 4) {
        v2f a, bb;
        a.x = As[l16][k + kb];
        a.y = As[l16][k + kb + 1];
        int p0 = k + kb;     p0 = p0 < NPF ? p0 : NPF-1;   // row 207 pairs with A==0
        int p1 = k + kb + 1; p1 = p1 < NPF ? p1 : NPF-1;
        bb.x = posedirs[p0*NCOL + nc];
        bb.y = posedirs[p1*NCOL + nc];
        acc = __builtin_amdgcn_wmma_f32_16x16x4_f32(false, a, false, bb,
                                                    (short)0, acc, false, false);
    }
    if (n < NCOL) {
        #pragma unroll
        for (int r = 0; r < 8; ++r) {
            int row = mtile*16 + r + half*8;   // C/D layout: M = r + 8*(lane>=16)
            int idx = row*NCOL + n;
            vsh[idx] = vsh[idx] + acc[r];
        }
    }
}

// ---------------- K6: transpose lbs_weights (V,J) -> (J,V)
__global__ void k_transpose_w(const float* __restrict__ w, float* __restrict__ wT)
{
    int o = blockIdx.x*blockDim.x + threadIdx.x;
    if (o >= NVERT*NJ) return;
    int j = o / NVERT, v = o % NVERT;
    wT[o] = w[v*NJ + j];
}

// ---------------- K7: skinning: verts = (sum_j w[v,j] A[b,j]) . [v_posed,1] + transl
__global__ void __launch_bounds__(256)
k_skin(const float* __restrict__ vposed,
       const float* __restrict__ wT,
       const float* __restrict__ Amat,
       const float* __restrict__ transl,
       float* __restrict__ verts)
{
    __shared__ float Ash[NJ*12];
    int b = blockIdx.y;
    for (int i = threadIdx.x; i < NJ*12; i += blockDim.x)
        Ash[i] = Amat[b*NJ*12 + i];
    __syncthreads();
    int v = blockIdx.x*blockDim.x + threadIdx.x;
    if (v >= NVERT) return;
    float T[12];
    #pragma unroll
    for (int i = 0; i < 12; ++i) T[i] = 0.f;
    #pragma unroll
    for (int j = 0; j < NJ; ++j) {
        float w = wT[j*NVERT + v];
        #pragma unroll
        for (int i = 0; i < 12; ++i) T[i] += w*Ash[j*12+i];
    }
    float x = vposed[(b*NVERT+v)*3+0];
    float y = vposed[(b*NVERT+v)*3+1];
    float z = vposed[(b*NVERT+v)*3+2];
    #pragma unroll
    for (int m = 0; m < 3; ++m) {
        float o = T[m*4+0]*x + T[m*4+1]*y + T[m*4+2]*z + T[m*4+3] + transl[b*3+m];
        verts[(b*NVERT+v)*3+m] = o;
    }
}

extern "C" void kernel_launch(void* const* d_in, const int* in_sizes, int n_in,
                              void* d_out, int out_size, void* d_ws, size_t ws_size,
                              hipStream_t stream)
{
    const float* betas         = (const float*)d_in[0];
    const float* body_pose     = (const float*)d_in[1];
    const float* global_orient = (const float*)d_in[2];
    const float* transl        = (const float*)d_in[3];
    const float* v_template    = (const float*)d_in[4];
    const float* shapedirs     = (const float*)d_in[5];
    const float* posedirs      = (const float*)d_in[6];
    const float* J_regressor   = (const float*)d_in[7];
    const float* lbs_weights   = (const float*)d_in[8];

    float* out_verts  = (float*)d_out;                       // [B][V][3]
    float* out_joints = out_verts + (size_t)NBATCH*NVERT*3;  // [B][24][3]

    // workspace layout (floats)
    float* ws = (float*)d_ws;
    float* ws_rot   = ws;                                  // B*24*9
    float* ws_pf    = ws_rot   + (size_t)NBATCH*NJ*9;      // B*207
    float* ws_vsh   = ws_pf    + (size_t)NBATCH*NPF;       // B*V*3 (becomes v_posed)
    float* ws_jrest = ws_vsh   + (size_t)NBATCH*NVERT*3;   // B*24*3
    float* ws_A     = ws_jrest + (size_t)NBATCH*NJ*3;      // B*24*12
    float* ws_wT    = ws_A     + (size_t)NBATCH*NJ*12;     // 24*V

    k_rodrigues<<<NBATCH, 32, 0, stream>>>(body_pose, global_orient, ws_rot, ws_pf);

    dim3 gV((NVERT + 255)/256, NBATCH);
    k_vshaped<<<gV, 256, 0, stream>>>(betas, v_template, shapedirs, ws_vsh);

    k_jrest<<<NBATCH, 256, 0, stream>>>(J_regressor, ws_vsh, ws_jrest);

    k_chain<<<NBATCH, 32, 0, stream>>>(ws_rot, ws_jrest, transl, ws_A, out_joints);

    k_transpose_w<<<(NVERT*NJ + 255)/256, 256, 0, stream>>>(lbs_weights, ws_wT);

    // N tiles: ceil(20670/16)=1292; 4 waves/block -> 323 blocks in x; 32 M tiles
    dim3 gP(323, 32);
    k_poseblend<<<gP, 128, 0, stream>>>(ws_pf, posedirs, ws_vsh);

    k_skin<<<gV, 256, 0, stream>>>(ws_vsh, ws_wT, ws_A, transl, out_verts);
}